// UNetTransformerWithAttentionFusion_27702539059240
// MI455X (gfx1250) — compile-verified
//
#include <hip/hip_runtime.h>
#include <math.h>

typedef __attribute__((ext_vector_type(2))) float v2f;
typedef __attribute__((ext_vector_type(8))) float v8f;

#define N_TOK 4096
#define C_DIM 64
#define NHEAD 8
#define DHEAD 8
#define TOPK  16
#define EPSLN 1e-5f
#define KCHUNK 512

__device__ __forceinline__ float waveMax(float v) {
#pragma unroll
  for (int m = 16; m >= 1; m >>= 1) v = fmaxf(v, __shfl_xor(v, m, 32));
  return v;
}
__device__ __forceinline__ float waveSum(float v) {
#pragma unroll
  for (int m = 16; m >= 1; m >>= 1) v += __shfl_xor(v, m, 32);
  return v;
}

// -------- LayerNorm over C=64: one wave per row, 2 channels per lane --------
__global__ __launch_bounds__(128) void ln_kernel(const float* __restrict__ x,
                                                 const float* __restrict__ w,
                                                 const float* __restrict__ b,
                                                 float* __restrict__ y) {
  int wave = threadIdx.x >> 5, lane = threadIdx.x & 31;
  int row = blockIdx.x * 4 + wave;
  const float* xr = x + row * C_DIM;
  float a0 = xr[lane], a1 = xr[lane + 32];
  float mu = waveSum(a0 + a1) * (1.0f / 64.0f);
  float d0 = a0 - mu, d1 = a1 - mu;
  float var = waveSum(d0 * d0 + d1 * d1) * (1.0f / 64.0f);
  float rs = rsqrtf(var + EPSLN);
  float* yr = y + row * C_DIM;
  yr[lane]      = d0 * rs * w[lane]      + b[lane];
  yr[lane + 32] = d1 * rs * w[lane + 32] + b[lane + 32];
}

// -------- Generic fp32 WMMA GEMM: out[M,N] = act(A[M,K] @ W[N,K]^T + bias) (+res)
// One wave -> one 16x16 tile via V_WMMA_F32_16X16X4_F32 (K stepped by 4).
// A-lane layout: a.x=A[m][k+2*(lane>=16)], a.y = next K.  B mirrored from W rows.
template <bool RELU, bool RES>
__global__ __launch_bounds__(128) void gemm_wmma(const float* __restrict__ A,
                                                 const float* __restrict__ W,
                                                 const float* __restrict__ bias,
                                                 const float* __restrict__ res,
                                                 float* __restrict__ out,
                                                 int M, int N, int K) {
  int wave = threadIdx.x >> 5, lane = threadIdx.x & 31;
  int tile = blockIdx.x * 4 + wave;
  int ntiles = N >> 4;
  int m0 = (tile / ntiles) << 4, n0 = (tile % ntiles) << 4;
  int l16 = lane & 15, hi = lane >> 4;
  const float* arow = A + (size_t)(m0 + l16) * K + 2 * hi;
  const float* brow = W + (size_t)(n0 + l16) * K + 2 * hi;
  v8f c = {};
  for (int k = 0; k < K; k += 4) {
    v2f a = *(const v2f*)(arow + k);
    v2f b = *(const v2f*)(brow + k);
    c = __builtin_amdgcn_wmma_f32_16x16x4_f32(false, a, false, b, (short)0, c,
                                              false, false);
  }
  int n = n0 + l16;
  float bv = bias[n];
#pragma unroll
  for (int i = 0; i < 8; ++i) {
    int m = m0 + i + 8 * hi;
    float v = c[i] + bv;
    if (RELU) v = fmaxf(v, 0.0f);
    if (RES) v += res[(size_t)m * N + n];
    out[(size_t)m * N + n] = v;
  }
}

// -------- Per-head V column sums: vsum[64] over 4096 keys --------
__global__ __launch_bounds__(256) void vsum_kernel(const float* __restrict__ kv,
                                                   float* __restrict__ vsum) {
  __shared__ float red[256];
  int d = blockIdx.x;  // 0..63
  float s = 0.f;
  for (int m = threadIdx.x; m < N_TOK; m += 256) s += kv[m * 128 + 64 + d];
  red[threadIdx.x] = s;
  __syncthreads();
  for (int o = 128; o > 0; o >>= 1) {
    if (threadIdx.x < (unsigned)o) red[threadIdx.x] += red[threadIdx.x + o];
    __syncthreads();
  }
  if (threadIdx.x == 0) vsum[d] = red[0];
}

// -------- Fused sparse attention core: scores + top-16 + masked softmax + AV
// Block = 256 thr = 8 waves; wave owns one (head,row); K staged in LDS (stride 9).
__global__ __launch_bounds__(256) void attn_kernel(const float* __restrict__ q,
                                                   const float* __restrict__ kv,
                                                   const float* __restrict__ vsum,
                                                   float* __restrict__ outp) {
  __shared__ float lk[KCHUNK * 9];
  int wave = threadIdx.x >> 5, lane = threadIdx.x & 31;
  int head = blockIdx.x >> 9;
  int row = ((blockIdx.x & 511) << 3) + wave;
  float qreg[8];
#pragma unroll
  for (int j = 0; j < 8; ++j) qreg[j] = q[row * 64 + head * 8 + j];
  const float scale = 0.35355339059327373f;  // 8^-0.5
  float val[16];
  int idx[16];
#pragma unroll
  for (int i = 0; i < 16; ++i) { val[i] = -INFINITY; idx[i] = 0; }

  for (int c0 = 0; c0 < N_TOK; c0 += KCHUNK) {
    for (int t = threadIdx.x; t < KCHUNK * 8; t += 256) {
      int ml = t >> 3, j = t & 7;
      lk[ml * 9 + j] = kv[(size_t)(c0 + ml) * 128 + head * 8 + j];
    }
    __syncthreads();
#pragma unroll 4
    for (int t = 0; t < KCHUNK / 32; ++t) {
      int ml = t * 32 + lane;
      const float* kr = &lk[ml * 9];
      float s = qreg[0] * kr[0];
#pragma unroll
      for (int j = 1; j < 8; ++j) s += qreg[j] * kr[j];
      s *= scale;
      if (s > val[15]) {  // register insertion sort, fully static indexing
        float cv = s; int ci = c0 + ml;
#pragma unroll
        for (int i = 0; i < 16; ++i) {
          bool gt = cv > val[i];
          float tv = val[i]; int ti = idx[i];
          val[i] = gt ? cv : tv; idx[i] = gt ? ci : ti;
          cv = gt ? tv : cv;   ci = gt ? ti : ci;
        }
      }
    }
    __syncthreads();
  }

  // Global max -> M = max(s_max, 0); pop global top-16 while accumulating.
  float rowmax = waveMax(val[0]);
  float Mx = fmaxf(rowmax, 0.0f);
  float e0 = expf(-Mx);
  float z = 0.f, acc[8] = {0, 0, 0, 0, 0, 0, 0, 0};
  for (int it = 0; it < TOPK; ++it) {
    float m = waveMax(val[0]);
    if (val[0] == m && m > -INFINITY) {
      float wgt = expf(val[0] - Mx) - e0;
      const float* vr = kv + (size_t)idx[0] * 128 + 64 + head * 8;
      z += wgt;
#pragma unroll
      for (int j = 0; j < 8; ++j) acc[j] += wgt * vr[j];
#pragma unroll
      for (int i = 0; i < 15; ++i) { val[i] = val[i + 1]; idx[i] = idx[i + 1]; }
      val[15] = -INFINITY;
    }
  }
  z = waveSum(z);
#pragma unroll
  for (int j = 0; j < 8; ++j) acc[j] = waveSum(acc[j]);
  if (lane == 0) {
    float inv = 1.0f / (z + (float)N_TOK * e0);
#pragma unroll
    for (int j = 0; j < 8; ++j)
      outp[row * 64 + head * 8 + j] = (acc[j] + e0 * vsum[head * 8 + j]) * inv;
  }
}

// -------- grouped 3x3 conv (groups=8) + fuse-softmax + LN(norm1) + residual ---
__global__ __launch_bounds__(256) void local_fuse_kernel(
    const float* __restrict__ x, const float* __restrict__ g,
    const float* __restrict__ lw, const float* __restrict__ lb,
    const float* __restrict__ fw, const float* __restrict__ n1w,
    const float* __restrict__ n1b, float* __restrict__ x1) {
  int wave = threadIdx.x >> 5, lane = threadIdx.x & 31;
  int p = blockIdx.x * 8 + wave;
  int h = p >> 6, w = p & 63;
  float ef0 = expf(fw[0]), ef1 = expf(fw[1]);
  float w0 = ef0 / (ef0 + ef1), w1 = ef1 / (ef0 + ef1);
  float t[2];
#pragma unroll
  for (int half = 0; half < 2; ++half) {
    int oc = lane + 32 * half;
    int gr = oc >> 3;
    float acc = lb[oc];
#pragma unroll
    for (int dh = -1; dh <= 1; ++dh) {
      int hh = h + dh;
#pragma unroll
      for (int dw = -1; dw <= 1; ++dw) {
        int ww = w + dw;
        bool in = (hh >= 0 && hh < 64 && ww >= 0 && ww < 64);
        int pp = hh * 64 + ww;
#pragma unroll
        for (int ic = 0; ic < 8; ++ic) {
          float xv = in ? x[(size_t)pp * 64 + gr * 8 + ic] : 0.f;
          acc += xv * lw[((oc * 8 + ic) * 3 + (dh + 1)) * 3 + (dw + 1)];
        }
      }
    }
    t[half] = w0 * g[(size_t)p * 64 + oc] + w1 * fmaxf(acc, 0.f);
  }
  float mu = waveSum(t[0] + t[1]) * (1.f / 64.f);
  float d0 = t[0] - mu, d1 = t[1] - mu;
  float var = waveSum(d0 * d0 + d1 * d1) * (1.f / 64.f);
  float rs = rsqrtf(var + EPSLN);
  x1[(size_t)p * 64 + lane]      = x[(size_t)p * 64 + lane]      + d0 * rs * n1w[lane]      + n1b[lane];
  x1[(size_t)p * 64 + lane + 32] = x[(size_t)p * 64 + lane + 32] + d1 * rs * n1w[lane + 32] + n1b[lane + 32];
}

// -------- MSFN depthwise 3x3 & 5x5 (1->4 per group), shared 5x5 tap regs -----
__global__ __launch_bounds__(256) void msfn_conv_kernel(
    const float* __restrict__ ln2x, const float* __restrict__ w3,
    const float* __restrict__ b3, const float* __restrict__ w5,
    const float* __restrict__ b5, float* __restrict__ y) {
  int gid = blockIdx.x * 256 + threadIdx.x;  // 4096*64
  int p = gid >> 6, gch = gid & 63;
  int h = p >> 6, w = p & 63;
  float tap[5][5];
#pragma unroll
  for (int dh = -2; dh <= 2; ++dh)
#pragma unroll
    for (int dw = -2; dw <= 2; ++dw) {
      int hh = h + dh, ww = w + dw;
      bool in = (hh >= 0 && hh < 64 && ww >= 0 && ww < 64);
      tap[dh + 2][dw + 2] = in ? ln2x[(size_t)(hh * 64 + ww) * 64 + gch] : 0.f;
    }
#pragma unroll
  for (int j = 0; j < 4; ++j) {
    int oc = gch * 4 + j;
    float a3 = b3[oc];
#pragma unroll
    for (int kh = 0; kh < 3; ++kh)
#pragma unroll
      for (int kw = 0; kw < 3; ++kw)
        a3 += tap[kh + 1][kw + 1] * w3[oc * 9 + kh * 3 + kw];
    y[(size_t)p * 512 + oc] = fmaxf(a3, 0.f);
    float a5 = b5[oc];
#pragma unroll
    for (int kh = 0; kh < 5; ++kh)
#pragma unroll
      for (int kw = 0; kw < 5; ++kw)
        a5 += tap[kh][kw] * w5[oc * 25 + kh * 5 + kw];
    y[(size_t)p * 512 + 256 + oc] = fmaxf(a5, 0.f);
  }
}

extern "C" void kernel_launch(void* const* d_in, const int* in_sizes, int n_in,
                              void* d_out, int out_size, void* d_ws, size_t ws_size,
                              hipStream_t stream) {
  const float* x   = (const float*)d_in[0];
  const float* n1w = (const float*)d_in[3];
  const float* n1b = (const float*)d_in[4];
  const float* qw  = (const float*)d_in[5];
  const float* qb  = (const float*)d_in[6];
  const float* kvw = (const float*)d_in[7];
  const float* kvb = (const float*)d_in[8];
  const float* pw  = (const float*)d_in[9];
  const float* pb  = (const float*)d_in[10];
  const float* lw  = (const float*)d_in[11];
  const float* lb  = (const float*)d_in[12];
  const float* fw  = (const float*)d_in[13];
  const float* n2w = (const float*)d_in[14];
  const float* n2b = (const float*)d_in[15];
  const float* w3  = (const float*)d_in[16];
  const float* b3  = (const float*)d_in[17];
  const float* w5  = (const float*)d_in[18];
  const float* b5  = (const float*)d_in[19];
  const float* c1w = (const float*)d_in[20];
  const float* c1b = (const float*)d_in[21];
  float* out = (float*)d_out;

  float* ws      = (float*)d_ws;
  float* ln1x    = ws;             // 262144
  float* q       = ws + 262144;    // 262144
  float* kv      = ws + 524288;    // 524288
  float* attnpre = ws + 1048576;   // 262144
  float* g       = ws + 1310720;   // 262144
  float* x1      = ws + 1572864;   // 262144
  float* ln2x    = ws + 1835008;   // 262144
  float* y       = ws + 2097152;   // 2097152
  float* vsum    = ws + 4194304;   // 64

  ln_kernel<<<1024, 128, 0, stream>>>(x, n1w, n1b, ln1x);
  gemm_wmma<true,  false><<<256, 128, 0, stream>>>(ln1x, qw,  qb,  nullptr, q,  N_TOK,  64, 64);
  gemm_wmma<false, false><<<512, 128, 0, stream>>>(ln1x, kvw, kvb, nullptr, kv, N_TOK, 128, 64);
  vsum_kernel<<<64, 256, 0, stream>>>(kv, vsum);
  attn_kernel<<<4096, 256, 0, stream>>>(q, kv, vsum, attnpre);
  gemm_wmma<false, false><<<256, 128, 0, stream>>>(attnpre, pw, pb, nullptr, g, N_TOK, 64, 64);
  local_fuse_kernel<<<512, 256, 0, stream>>>(x, g, lw, lb, fw, n1w, n1b, x1);
  ln_kernel<<<1024, 128, 0, stream>>>(x1, n2w, n2b, ln2x);
  msfn_conv_kernel<<<1024, 256, 0, stream>>>(ln2x, w3, b3, w5, b5, y);
  gemm_wmma<false, true><<<256, 128, 0, stream>>>(y, c1w, c1b, x1, out, N_TOK, 64, 512);
}